// ConvEmbedder_65481071404682
// MI455X (gfx1250) — compile-verified
//
#include <hip/hip_runtime.h>

typedef __attribute__((ext_vector_type(2))) float v2f;
typedef __attribute__((ext_vector_type(8))) float v8f;

// Fixed problem shape from the reference.
#define BATCH 64
#define IMG   512
#define KSZ   16
#define NPROW 32                 // IMG / KSZ
#define PPB   (NPROW * NPROW)    // 1024 patches per image
#define NP    (BATCH * PPB)      // 65536 patches total
#define EMB   768

// ---------------------------------------------------------------------------
// Phase 1: v[patch] = dot(patch_16x16, conv_w) + conv_b
// One wave32 per patch; 8 elements per lane, coalesced 64B row segments,
// shfl-xor tree reduction. Reads data exactly once -> non-temporal loads.
// ---------------------------------------------------------------------------
__global__ __launch_bounds__(256) void patch_dot_kernel(
    const float* __restrict__ data, const float* __restrict__ cw,
    const float* __restrict__ cb, float* __restrict__ v)
{
    const int lane  = threadIdx.x & 31;
    const int patch = (blockIdx.x * blockDim.x + threadIdx.x) >> 5;  // 0..NP-1
    const int b  = patch >> 10;          // / PPB
    const int p  = patch & (PPB - 1);
    const int ph = p >> 5;               // / NPROW
    const int pw = p & (NPROW - 1);

    const float* base = data + (size_t)b * IMG * IMG
                             + (size_t)(ph * KSZ) * IMG + pw * KSZ;
    float acc = 0.0f;
#pragma unroll
    for (int i = 0; i < 8; ++i) {
        const int k = lane + i * 32;        // element index within patch
        const int r = k >> 4, c = k & 15;
        acc = fmaf(__builtin_nontemporal_load(base + r * IMG + c), cw[k], acc);
    }
#pragma unroll
    for (int off = 16; off > 0; off >>= 1)
        acc += __shfl_xor(acc, off, 32);
    if (lane == 0) v[patch] = acc + cb[0];
}

// ---------------------------------------------------------------------------
// Phase 2: out[p, e] = v[p] * lin_w[e] + lin_b[e] as rank-1 WMMA tiles.
// One wave per 16 patches; 48 embed tiles of 16 via V_WMMA_F32_16X16X4_F32:
//   A[m,0] = v[m]      (A vgpr0, lanes 0-15; everything else 0)
//   B[0,n] = lin_w[n]  (B vgpr0, lanes 0-15; everything else 0)
//   C[m,n] = lin_b[n]  (every C vgpr = lin_b[e0 + lane%16])
// D vgpr i, lane L -> row (i + 8*(L/16)), col (L%16): stored as 8 b32 stores,
// each two contiguous 64B segments. Output > L2 -> non-temporal stores.
// ---------------------------------------------------------------------------
__global__ __launch_bounds__(256) void embed_wmma_kernel(
    const float* __restrict__ v, const float* __restrict__ lw,
    const float* __restrict__ lb, float* __restrict__ out)
{
    const int lane = threadIdx.x & 31;
    const int tile = (blockIdx.x * blockDim.x + threadIdx.x) >> 5;  // 16 patches
    const int half = lane >> 4;    // 0: lanes 0-15, 1: lanes 16-31
    const int l16  = lane & 15;

    // A matrix: only column K=0 is non-zero.
    const float vm = v[tile * 16 + l16];
    v2f a;
    a.x = half ? 0.0f : vm;   // vgpr0: K=0 (lanes 0-15) / K=2 (lanes 16-31)
    a.y = 0.0f;               // vgpr1: K=1 / K=3

    float* outb = out + (size_t)tile * 16 * EMB;

    for (int e0 = 0; e0 < EMB; e0 += 16) {
        // B matrix: only row K=0 is non-zero.
        const float wv = lw[e0 + l16];
        v2f bm;
        bm.x = half ? 0.0f : wv;
        bm.y = 0.0f;

        // C matrix: bias broadcast along rows.
        const float bias = lb[e0 + l16];
        v8f c;
#pragma unroll
        for (int i = 0; i < 8; ++i) c[i] = bias;

        v8f d = __builtin_amdgcn_wmma_f32_16x16x4_f32(
            /*neg_a=*/false, a, /*neg_b=*/false, bm,
            /*c_mod=*/(short)0, c, /*reuse_a=*/false, /*reuse_b=*/false);

#pragma unroll
        for (int i = 0; i < 8; ++i) {
            __builtin_nontemporal_store(
                d[i], outb + (size_t)(i + 8 * half) * EMB + (e0 + l16));
        }
    }
}

extern "C" void kernel_launch(void* const* d_in, const int* in_sizes, int n_in,
                              void* d_out, int out_size, void* d_ws, size_t ws_size,
                              hipStream_t stream) {
    const float* data = (const float*)d_in[0];   // [64,512,512]
    const float* cw   = (const float*)d_in[1];   // [16,16]
    const float* cb   = (const float*)d_in[2];   // [1]
    const float* lw   = (const float*)d_in[3];   // [768]
    const float* lb   = (const float*)d_in[4];   // [768]
    float* out  = (float*)d_out;                 // [64,1024,768]
    float* vbuf = (float*)d_ws;                  // NP floats = 256 KB scratch

    // Phase 1: 65536 patches, 8 waves (8 patches) per 256-thread block.
    patch_dot_kernel<<<NP / 8, 256, 0, stream>>>(data, cw, cb, vbuf);
    // Phase 2: 4096 patch-tiles, 8 waves (8 tiles) per 256-thread block.
    embed_wmma_kernel<<<(NP / 16) / 8, 256, 0, stream>>>(vbuf, lw, lb, out);
}